// token_selection_28853590294612
// MI455X (gfx1250) — compile-verified
//
#include <hip/hip_runtime.h>
#include <stdint.h>

// MI455X / gfx1250 token-selection pipeline.
// B=32, N=4096, D=256, K=ceil(N/2)=2048.
// Workspace layout (needs 134,512,640 bytes):
//   [0, 128MB)          : scores [B][256][N] f32, later overwritten in-place by u32 indices
//   [+134217728, +256KB): M fused weight [256][256]
//   [+262144,  +32KB)   : c constants [B][256]

typedef __attribute__((ext_vector_type(2))) float v2f;
typedef __attribute__((ext_vector_type(4))) float v4f;
typedef __attribute__((ext_vector_type(8))) float v8f;

typedef __attribute__((address_space(1))) int g_i32;  // global-mem int
typedef __attribute__((address_space(3))) int l_i32;  // LDS int

#define TS_B 32
#define TS_N 4096
#define TS_D 256
#define TS_K 2048

#if defined(__gfx1250__) && __has_builtin(__builtin_amdgcn_global_load_async_to_lds_b32)
#define TS_HAVE_ASYNC_LDS 1
#else
#define TS_HAVE_ASYNC_LDS 0
#endif

__device__ __forceinline__ void ts_wait_asynccnt0() {
#if __has_builtin(__builtin_amdgcn_s_wait_asynccnt)
  __builtin_amdgcn_s_wait_asynccnt(0);
#else
  asm volatile("s_wait_asynccnt 0x0" ::: "memory");
#endif
}

// ---------------------------------------------------------------------------
// M[e][f] = sum_h Wf[h][e] * Ws[f][128+h]   (fused second-half weight)
// ---------------------------------------------------------------------------
__global__ __launch_bounds__(256) void ts_prep_m(const float* __restrict__ Wf,
                                                 const float* __restrict__ Ws,
                                                 float* __restrict__ M) {
  __shared__ float col[128];
  const int e = blockIdx.x;
  const int f = threadIdx.x;
  if (f < 128) col[f] = Wf[f * 256 + e];
  __syncthreads();
  float s = 0.f;
#pragma unroll 8
  for (int h = 0; h < 128; ++h) s = fmaf(col[h], Ws[f * 256 + 128 + h], s);
  M[e * 256 + f] = s;
}

// ---------------------------------------------------------------------------
// c[b][f] = sum_h (cls[b]·Wf[h] + bf[h])·Ws[f][h] + sum_h bf[h]·Ws[f][128+h] + bs[f]
// (constant in n -> cannot change per-(b,f) top-k order, kept for fidelity)
// ---------------------------------------------------------------------------
__global__ __launch_bounds__(256) void ts_prep_c(const float* __restrict__ cls,
                                                 const float* __restrict__ Wf,
                                                 const float* __restrict__ bf,
                                                 const float* __restrict__ Ws,
                                                 const float* __restrict__ bs,
                                                 float* __restrict__ C) {
  __shared__ float hc[128];
  const int b = blockIdx.x;
  const int f = threadIdx.x;
  if (f < 128) {
    float s = bf[f];
#pragma unroll 8
    for (int e = 0; e < 256; ++e) s = fmaf(cls[b * 256 + e], Wf[f * 256 + e], s);
    hc[f] = s;
  }
  __syncthreads();
  float s = bs[f];
#pragma unroll 8
  for (int h = 0; h < 128; ++h) {
    s = fmaf(hc[h], Ws[f * 256 + h], s);
    s = fmaf(bf[h], Ws[f * 256 + 128 + h], s);
  }
  C[b * 256 + f] = s;
}

// ---------------------------------------------------------------------------
// Main GEMM: score[b,n,f] = X[b,n,:]·M[:,f] + c[b,f], stored as S[b][f][n].
// Grid: 2048 blocks of 64 rows; 128 threads = 4 waves; each wave: 16 rows x 256 f
// via V_WMMA_F32_16X16X4_F32. M k-chunks (4x256 = contiguous 4KB) staged into
// LDS in transposed [f][kk] layout via GLOBAL_LOAD_ASYNC_TO_LDS_B32
// (ASYNCcnt-tracked copy engine, per-lane LDS addresses do the transpose).
// ---------------------------------------------------------------------------
__global__ __launch_bounds__(128) void ts_gemm(const float* __restrict__ X,
                                               const float* __restrict__ M,
                                               const float* __restrict__ C,
                                               float* __restrict__ S) {
  __shared__ float lm[1024];  // M chunk, [f][kk] kk=0..3 -> conflict-free ds_load_2addr_b64
  const int tid   = threadIdx.x;
  const int lane  = tid & 31;
  const int wave  = tid >> 5;
  const int fcol  = lane & 15;          // channel within a 16-wide f tile
  const int khalf = (lane >> 4) << 1;   // 0 or 2 (K split across lane halves)
  const int row0  = blockIdx.x * 64 + wave * 16;
  const int b     = row0 >> 12;         // 4096 rows per batch
  const int nb    = row0 & (TS_N - 1);
  const int arow  = row0 + fcol;

  v8f acc[16];
#pragma unroll
  for (int t = 0; t < 16; ++t) {
    const float cc = C[b * 256 + t * 16 + fcol];
    acc[t] = (v8f){cc, cc, cc, cc, cc, cc, cc, cc};
  }

  for (int k = 0; k < 256; k += 4) {
    __syncthreads();  // previous chunk fully consumed by all waves
#if TS_HAVE_ASYNC_LDS
#pragma unroll
    for (int q = 0; q < 8; ++q) {
      const int i = tid + q * 128;  // chunk is contiguous: M[k*256 + i]
      __builtin_amdgcn_global_load_async_to_lds_b32(
          (g_i32*)(M + k * 256 + i),
          (l_i32*)&lm[(i & 255) * 4 + (i >> 8)],
          /*offset=*/0, /*cpol=*/0);
    }
    ts_wait_asynccnt0();
#else
    float stage[8];
#pragma unroll
    for (int q = 0; q < 8; ++q) stage[q] = M[k * 256 + tid + q * 128];  // 8 in-flight loads
#pragma unroll
    for (int q = 0; q < 8; ++q) {
      const int i = tid + q * 128;
      lm[(i & 255) * 4 + (i >> 8)] = stage[q];
    }
#endif
    __syncthreads();
    const v2f a = *(const v2f*)(X + (size_t)arow * 256 + k + khalf);
#pragma unroll
    for (int t = 0; t < 16; ++t) {
      const v2f bb = *(const v2f*)(lm + (t * 16 + fcol) * 4 + khalf);
      acc[t] = __builtin_amdgcn_wmma_f32_16x16x4_f32(
          /*neg_a=*/false, a, /*neg_b=*/false, bb,
          /*c_mod=*/(short)0, acc[t], /*reuse_a=*/false, /*reuse_b=*/false);
    }
  }

  // Transposed store: lane owns 8 consecutive-n values of one channel f.
#pragma unroll
  for (int t = 0; t < 16; ++t) {
    const int f = t * 16 + fcol;
    float* p = S + (((size_t)(b * 256 + f)) << 12) + nb + ((lane >> 4) << 3);
    const v4f lo = {acc[t][0], acc[t][1], acc[t][2], acc[t][3]};
    const v4f hi = {acc[t][4], acc[t][5], acc[t][6], acc[t][7]};
    *(v4f*)p       = lo;
    *(v4f*)(p + 4) = hi;
  }
}

// ---------------------------------------------------------------------------
// Per (b,f): sorted top-2048 indices of 4096 scores (descending value,
// ascending index ties -> matches top_k). Bitonic sort of u64 keys in LDS.
// Indices written IN PLACE over this row's score region (safe: row fully
// loaded into LDS before any write; no other WG touches this row).
// ---------------------------------------------------------------------------
__global__ __launch_bounds__(1024) void ts_sort(float* __restrict__ S) {
  __shared__ unsigned long long key[TS_N];  // 32 KB
  const int tid = threadIdx.x;
  const unsigned base = (unsigned)blockIdx.x << 12;  // (b*256+f)*4096

  for (int i = tid; i < TS_N; i += 1024) {
    const unsigned u = __float_as_uint(S[base + i]);
    const unsigned mono = (u & 0x80000000u) ? ~u : (u | 0x80000000u);  // asc map
    key[i] = (((unsigned long long)(~mono)) << 32) | (unsigned)i;       // asc key = desc val
  }
  for (unsigned k = 2; k <= TS_N; k <<= 1) {
    for (unsigned j = k >> 1; j > 0; j >>= 1) {
      __syncthreads();
      for (unsigned v = (unsigned)tid; v < TS_N / 2; v += 1024) {
        const unsigned i = ((v & ~(j - 1)) << 1) | (v & (j - 1));
        const unsigned l = i | j;
        const unsigned long long a = key[i], c = key[l];
        if ((a > c) == ((i & k) == 0)) { key[i] = c; key[l] = a; }
      }
    }
  }
  __syncthreads();
  unsigned* idx = (unsigned*)S;
  for (int i = tid; i < TS_K; i += 1024)
    idx[base + i] = (unsigned)(key[i] & 0xFFFFFFFFu);
}

// ---------------------------------------------------------------------------
// out[b][j][f] = X[b][ idx[b][f][j] ][f]; contiguous writes, L2-served reads.
// ---------------------------------------------------------------------------
__global__ __launch_bounds__(256) void ts_gather(const float* __restrict__ X,
                                                 const unsigned* __restrict__ IDX,
                                                 float* __restrict__ O) {
  const int f = threadIdx.x;
  const int b = blockIdx.x >> 11;  // 2048 j-slots per batch
  const int j = blockIdx.x & (TS_K - 1);
  const unsigned n = IDX[(((unsigned)(b * 256 + f)) << 12) + j];
  O[((size_t)blockIdx.x << 8) + f] = X[(((size_t)(b * TS_N + (int)n)) << 8) + f];
}

// ---------------------------------------------------------------------------
extern "C" void kernel_launch(void* const* d_in, const int* in_sizes, int n_in,
                              void* d_out, int out_size, void* d_ws, size_t ws_size,
                              hipStream_t stream) {
  const float* X   = (const float*)d_in[0];  // [32,4096,256]
  const float* cls = (const float*)d_in[1];  // [32,256]
  const float* Wf  = (const float*)d_in[2];  // [128,256]
  const float* bf  = (const float*)d_in[3];  // [128]
  const float* Ws  = (const float*)d_in[4];  // [256,256]
  const float* bs  = (const float*)d_in[5];  // [256]
  float* out = (float*)d_out;                // [32,2048,256]

  char* ws = (char*)d_ws;
  float* scores = (float*)ws;                                  // 134,217,728 B
  float* Mbuf   = (float*)(ws + 134217728u);                   // 262,144 B
  float* cbuf   = (float*)(ws + 134217728u + 262144u);         // 32,768 B

  ts_prep_m<<<256, 256, 0, stream>>>(Wf, Ws, Mbuf);
  ts_prep_c<<<32, 256, 0, stream>>>(cls, Wf, bf, Ws, bs, cbuf);
  ts_gemm<<<2048, 128, 0, stream>>>(X, Mbuf, cbuf, scores);
  ts_sort<<<8192, 1024, 0, stream>>>(scores);
  ts_gather<<<TS_B * TS_K, 256, 0, stream>>>(X, (const unsigned*)scores, out);
}